// MDNLoss_43087111913738
// MI455X (gfx1250) — compile-verified
//
#include <hip/hip_runtime.h>
#include <math.h>

// MDN loss: B=64, S=256, K=16, D=16.
// inputs (setup_inputs order): pi[B,S,K] f32, mu[B,S,K,D] f32,
// full_cov[B,S,K,D,D] f32, y[B,S,D] f32. output: scalar f32.

#define B_ 64
#define S_ 256
#define K_ 16
#define D_ 16
#define NCOMP (B_ * S_ * K_)   // 262144 16x16 Cholesky problems
#define NPAIR (B_ * S_)        // 16384 (b,s) pairs

typedef __attribute__((ext_vector_type(2))) float v2f;
typedef __attribute__((ext_vector_type(8))) float v8f;

#define LDS_STRIDE 20          // 16x20 f32 tile: 16B-aligned rows, conflict-free columns
#define WAVES_PER_BLOCK 8

#if defined(__has_builtin)
#  if __has_builtin(__builtin_amdgcn_global_load_async_to_lds_b128)
#    define HAVE_ASYNC_B128 1
#  endif
#  if __has_builtin(__builtin_amdgcn_s_wait_asynccnt)
#    define HAVE_WAIT_ASYNC 1
#  endif
#endif

#if HAVE_ASYNC_B128
// Builtin prototype (from clang diagnostic): param0 = int4(vector_size) in
// __device__ (AS1), param1 = int4 in __shared__ (AS3), then imm offset, imm cpol.
typedef int v4i_b __attribute__((vector_size(16)));
typedef __attribute__((address_space(1))) v4i_b gv4i;
typedef __attribute__((address_space(3))) v4i_b lv4i;
#endif

// Wave-local LDS fence: waves in a block own disjoint LDS tiles, so no
// s_barrier is needed. Within one wave32 (lockstep, LDS ops in-order per
// wave) a dscnt wait + compiler barrier orders cross-lane LDS handoff.
__device__ __forceinline__ void wave_lds_fence() {
    asm volatile("s_wait_dscnt 0x0" ::: "memory");
}

// One wave32 per (b,s,k) component: stage 16x16 cov in LDS (async-to-LDS
// when available), blocked Cholesky (panel width 4) with
// V_WMMA_F32_16X16X4_F32 rank-4 trailing updates, then wave-cooperative
// forward solve + log-det.
__global__ __launch_bounds__(256) void mdn_chol_kernel(
    const float* __restrict__ mu, const float* __restrict__ cov,
    const float* __restrict__ y, float* __restrict__ lp)
{
    __shared__ float smem[WAVES_PER_BLOCK * 16 * LDS_STRIDE];
    const int lane = threadIdx.x & 31;
    const int wave = threadIdx.x >> 5;
    const int comp = blockIdx.x * WAVES_PER_BLOCK + wave;
    float* A = smem + wave * 16 * LDS_STRIDE;

    const float* cw = cov + (size_t)comp * 256;
    // Stream-ahead prefetch of the next block's covariance tiles (global_prefetch_b8).
    __builtin_prefetch(cw + 256 * WAVES_PER_BLOCK, 0, 1);

    // Stage 16x16 cov into LDS [16][20]: each lane owns 8 consecutive floats
    // (two B128 transfers), all 16B-aligned in both global and LDS.
    {
        const int e0 = lane * 8;
        const int r = e0 >> 4;
        const int c = e0 & 15;           // 0 or 8
        float* dst = A + r * LDS_STRIDE + c;
#if HAVE_ASYNC_B128
        __builtin_amdgcn_global_load_async_to_lds_b128(
            (gv4i*)(cw + e0),     (lv4i*)dst,       0, 0);
        __builtin_amdgcn_global_load_async_to_lds_b128(
            (gv4i*)(cw + e0 + 4), (lv4i*)(dst + 4), 0, 0);
#  if HAVE_WAIT_ASYNC
        __builtin_amdgcn_s_wait_asynccnt(0);
#  else
        asm volatile("s_wait_asynccnt 0x0" ::: "memory");
#  endif
#else
        const float4* cv4 = (const float4*)cw;
        float4 p0 = cv4[lane * 2 + 0];
        float4 p1 = cv4[lane * 2 + 1];
        *(float4*)dst       = p0;
        *(float4*)(dst + 4) = p1;
#endif
    }
    wave_lds_fence();

    // Blocked right-looking Cholesky, panel width 4.
    for (int kb = 0; kb < 4; ++kb) {
        const int j0 = kb * 4;
        // ---- factor panel columns j0..j0+3 (rows handled by lanes 0..15) ----
        for (int jj = 0; jj < 4; ++jj) {
            const int j = j0 + jj;
            wave_lds_fence();
            const float djj = A[j * LDS_STRIDE + j];  // read before overwrite
            const float ljj = __builtin_amdgcn_sqrtf(djj);
            if (lane == j) A[j * LDS_STRIDE + j] = ljj;
            const float rinv = __builtin_amdgcn_rcpf(ljj);
            const int i = lane;
            float lij = 0.0f;
            if (i > j && i < 16) {
                lij = A[i * LDS_STRIDE + j] * rinv;
                A[i * LDS_STRIDE + j] = lij;
            }
            wave_lds_fence();
            // trailing update restricted to the remaining panel columns
            for (int c2 = j + 1; c2 < j0 + 4; ++c2) {
                const float lcj = A[c2 * LDS_STRIDE + j];
                if (i >= c2 && i < 16) A[i * LDS_STRIDE + c2] -= lij * lcj;
            }
        }
        // ---- WMMA rank-4 trailing update: T -= P * P^T ----
        if (kb < 3) {
            wave_lds_fence();
            const int r0 = j0 + 4;       // first trailing row/col
            // A-fragment (16x4 f32): lanes 0-15 hold M=lane K={0,1},
            // lanes 16-31 hold M=lane-16 K={2,3}. Rows < r0 zeroed so the
            // full 16x16 WMMA only perturbs the trailing block.
            const int m  = lane & 15;
            const int kk = (lane >> 4) * 2;
            const bool live = (m >= r0);
            const float p0 = live ? A[m * LDS_STRIDE + j0 + kk]     : 0.0f;
            const float p1 = live ? A[m * LDS_STRIDE + j0 + kk + 1] : 0.0f;
            v2f afrag; afrag.x = -p0; afrag.y = -p1;   // negate A: D = C - P*P^T
            v2f bfrag; bfrag.x =  p0; bfrag.y =  p1;   // B = P^T: mirrored layout
                                                       // => same lane values as A
            // C-fragment: lane n=lane&15, VGPR v -> M = v + 8*(lane>=16)
            const int n = lane & 15;
            const int mbase = (lane >> 4) * 8;
            v8f cfrag;
            #pragma unroll
            for (int v = 0; v < 8; ++v)
                cfrag[v] = A[(mbase + v) * LDS_STRIDE + n];
            cfrag = __builtin_amdgcn_wmma_f32_16x16x4_f32(
                false, afrag, false, bfrag, (short)0, cfrag, false, false);
            #pragma unroll
            for (int v = 0; v < 8; ++v)
                A[(mbase + v) * LDS_STRIDE + n] = cfrag[v];
        }
    }
    wave_lds_fence();

    // half_logdet = sum_i log L[i][i] (wave reduction; lanes >=16 contribute 0)
    float t = 0.0f;
    if (lane < 16) t = logf(A[lane * LDS_STRIDE + lane]);
    #pragma unroll
    for (int off = 16; off > 0; off >>= 1) t += __shfl_xor(t, off, 32);
    const float half_logdet = t;

    // forward solve L z = (y - mu), quad = |z|^2 (lane i owns row i)
    const int b   = comp / (S_ * K_);
    const int rem = comp - b * (S_ * K_);
    const int s   = rem / K_;
    float w = 0.0f;
    if (lane < 16) {
        const float yv = y[(size_t)(b * S_ + s) * D_ + lane];
        const float mv = mu[(size_t)comp * D_ + lane];
        w = yv - mv;
    }
    float quad = 0.0f;
    #pragma unroll
    for (int j = 0; j < 16; ++j) {
        const float zj = __shfl(w, j, 32) *
                         __builtin_amdgcn_rcpf(A[j * LDS_STRIDE + j]);
        quad += zj * zj;
        if (lane > j && lane < 16) w -= A[lane * LDS_STRIDE + j] * zj;
    }
    if (lane == 0) {
        const float LOG2PI = 1.8378770664093453f;
        lp[comp] = -0.5f * (quad + 16.0f * LOG2PI) - half_logdet;
    }
}

// One thread per (b,s): logsumexp_k(log pi_k + lp_k) - log(sum_k pi_k),
// block-reduce sums into per-block partials.
__global__ __launch_bounds__(256) void mdn_lse_kernel(
    const float* __restrict__ pi, const float* __restrict__ lp,
    float* __restrict__ partial)
{
    __shared__ float red[256];
    const int p = blockIdx.x * 256 + threadIdx.x;
    const float* pv = pi + (size_t)p * K_;
    const float* lv = lp + (size_t)p * K_;
    float cl[16];
    float mx = -INFINITY, sumpi = 0.0f;
    #pragma unroll
    for (int k = 0; k < 16; ++k) {
        const float pk = pv[k];
        sumpi += pk;
        cl[k] = logf(pk) + lv[k];
        mx = fmaxf(mx, cl[k]);
    }
    float se = 0.0f;
    #pragma unroll
    for (int k = 0; k < 16; ++k) se += expf(cl[k] - mx);
    const float val = mx + logf(se) - logf(sumpi);

    red[threadIdx.x] = val;
    __syncthreads();
    for (int off = 128; off > 0; off >>= 1) {
        if (threadIdx.x < off) red[threadIdx.x] += red[threadIdx.x + off];
        __syncthreads();
    }
    if (threadIdx.x == 0) partial[blockIdx.x] = red[0];
}

__global__ __launch_bounds__(64) void mdn_final_kernel(
    const float* __restrict__ partial, float* __restrict__ out)
{
    __shared__ float red[64];
    red[threadIdx.x] = partial[threadIdx.x];
    __syncthreads();
    for (int off = 32; off > 0; off >>= 1) {
        if (threadIdx.x < off) red[threadIdx.x] += red[threadIdx.x + off];
        __syncthreads();
    }
    if (threadIdx.x == 0) out[0] = -red[0] / (float)NPAIR;
}

extern "C" void kernel_launch(void* const* d_in, const int* in_sizes, int n_in,
                              void* d_out, int out_size, void* d_ws, size_t ws_size,
                              hipStream_t stream) {
    const float* pi  = (const float*)d_in[0];
    const float* mu  = (const float*)d_in[1];
    const float* cov = (const float*)d_in[2];
    const float* y   = (const float*)d_in[3];

    float* lp      = (float*)d_ws;        // NCOMP floats (1 MB)
    float* partial = lp + NCOMP;          // 64 floats

    mdn_chol_kernel<<<NCOMP / WAVES_PER_BLOCK, 256, 0, stream>>>(mu, cov, y, lp);
    mdn_lse_kernel<<<NPAIR / 256, 256, 0, stream>>>(pi, lp, partial);
    mdn_final_kernel<<<1, 64, 0, stream>>>(partial, (float*)d_out);
}